// GESelfAttentionBlock_2834678415574
// MI455X (gfx1250) — compile-verified
//
#include <hip/hip_runtime.h>

// GE self-attention block for gfx1250 (MI455X).
//   - f32 WMMA (V_WMMA_F32_16X16X4_F32) for all pair-invariant GEMMs (exact f32,
//     fully dense tiles: K-scores 8x64x32, values 48x8x32 per channel).
//   - async global->LDS streaming (ASYNCcnt) of the dominant 328 MB parallel-
//     transport tensor, issued at wave start and overlapped with the x-gather.
//
// Dims: V=40000 vertices, M=32 neighbors, C=8 channels, N=8 rep, B=8 basis.
// d_ws layout (f32): [WQ | WK | W0 | W1a | W1b | W2a | W2b | W2c], each [C][N][N] = 512.

typedef __attribute__((ext_vector_type(2))) float v2f;
typedef __attribute__((ext_vector_type(8))) float v8f;

static __device__ __forceinline__ v8f f8_zero() {
  v8f z;
#pragma unroll
  for (int h = 0; h < 8; ++h) z[h] = 0.f;
  return z;
}
static __device__ __forceinline__ v2f f2_zero() {
  v2f z; z[0] = 0.f; z[1] = 0.f;
  return z;
}
static __device__ __forceinline__ float wave_sum32(float v) {
#pragma unroll
  for (int off = 16; off > 0; off >>= 1) v += __shfl_xor(v, off, 32);
  return v;
}

// ---------------------------------------------------------------------------
// Kernel 0: fuse coefficients with bases:  W[c,i,j] = sum_b coeff[c,b]*basis[b,(o,)i,j]
// ---------------------------------------------------------------------------
__global__ void ge_prep_weights(const float* __restrict__ rtr,  // [B,N,N]
                                const float* __restrict__ qc,   // [C,B]
                                const float* __restrict__ kc,   // [C,B]
                                const float* __restrict__ vb0,  // [B,N,N]
                                const float* __restrict__ vb1,  // [B,2,N,N]
                                const float* __restrict__ vb2,  // [B,3,N,N]
                                const float* __restrict__ zp,   // [C,B]
                                const float* __restrict__ fp,   // [C,B]
                                const float* __restrict__ sp,   // [C,B]
                                float* __restrict__ w) {
  int idx = blockIdx.x * blockDim.x + threadIdx.x;
  if (idx >= 4096) return;
  int t = idx >> 9, rem = idx & 511;
  int c = rem >> 6, ij = rem & 63;
  const float* coeff;
  const float* basis;
  int bs, off;
  switch (t) {
    case 0: coeff = qc; basis = rtr; bs = 64;  off = 0;   break;
    case 1: coeff = kc; basis = rtr; bs = 64;  off = 0;   break;
    case 2: coeff = zp; basis = vb0; bs = 64;  off = 0;   break;
    case 3: coeff = fp; basis = vb1; bs = 128; off = 0;   break;
    case 4: coeff = fp; basis = vb1; bs = 128; off = 64;  break;
    case 5: coeff = sp; basis = vb2; bs = 192; off = 0;   break;
    case 6: coeff = sp; basis = vb2; bs = 192; off = 64;  break;
    default: coeff = sp; basis = vb2; bs = 192; off = 128; break;
  }
  float a = 0.f;
#pragma unroll
  for (int b = 0; b < 8; ++b) a += coeff[c * 8 + b] * basis[b * bs + off + ij];
  w[idx] = a;
}

// ---------------------------------------------------------------------------
// Main kernel: 1 wave = 1 vertex, 1 lane = 1 neighbor (wave32).
// ---------------------------------------------------------------------------
#define WPB 2        // waves (vertices) per workgroup
#define PSTRIDE 68   // padded row stride (floats) for the async P staging buffer

__launch_bounds__(32 * WPB)
__global__ void ge_attn(const float* __restrict__ x,       // [V,C,N]
                        const int* __restrict__ nbrs,      // [V,M]
                        const int* __restrict__ mask,      // [V,M] (0/1)
                        const float* __restrict__ ptm,     // [V,M,N,N]
                        const float* __restrict__ rpu,     // [V,M,2]
                        const float* __restrict__ W,       // fused weights (d_ws)
                        float* __restrict__ out,           // [V,C,N]
                        int nv) {
  // Shared weights (per workgroup)
  __shared__ float WQf[512];            // [c][i][j] (query map)
  __shared__ float WKf[512];            // [i][c*8+j] : K-score GEMM A, K=64 dense
  __shared__ float Wvf[8 * 48 * 8];     // [c][r=8*t+i][j], t in {W0,W1a,W1b,W2a,W2b,W2c}
  // Per-wave scratch
  __shared__ float Ff[WPB][32 * 64];    // [wave][m][c*8+i] : transported features f
  // Scr is time-multiplexed: phase A = async P staging [32][PSTRIDE];
  // per-channel loop = value GEMM results Vd[48][33].
  __shared__ float Scr[WPB][32 * PSTRIDE];
  __shared__ float Kls[WPB][8 * 33];    // [wave][i][m] K-scores
  __shared__ float Qls[WPB][8];         // [wave][i]   Q

  const int tid  = threadIdx.x;
  const int wave = tid >> 5;
  const int lane = tid & 31;
  const int hi   = lane >> 4;   // wave half (WMMA f32 layout: K = 2*hi + elem)
  const int lo   = lane & 15;

  int v = blockIdx.x * WPB + wave;
  if (v >= nv) v = nv - 1;  // duplicate work on tail (identical writes)

  // ---- issue async copy of this vertex's 8 KB ptm slice into LDS -------------
  // Linear element p = t*128 + lane*4 of ptm[v] maps to padded LDS slot
  // (p>>6)*PSTRIDE + (p&63)  ->  lane dest = (2t + hi)*PSTRIDE + lo*4.
  {
    const float* gp = ptm + (size_t)v * 2048 + lane * 4;
    unsigned lbase = (unsigned)(unsigned long long)
        (&Scr[wave][(unsigned)hi * PSTRIDE + (unsigned)lo * 4]);
#pragma unroll
    for (int t = 0; t < 16; ++t) {
      asm volatile("global_load_async_to_lds_b128 %0, %1, off"
                   :: "v"(lbase + (unsigned)(t * 2 * PSTRIDE * 4)),
                      "v"(gp + t * 128)
                   : "memory");
    }
  }

  // ---- workgroup weight prep (overlaps with the async stream) ----------------
  for (int i = tid; i < 512; i += 32 * WPB) WQf[i] = W[i];
  for (int i = tid; i < 512; i += 32 * WPB) {
    int i8 = i >> 6, cj = i & 63, c = cj >> 3, j = cj & 7;
    WKf[i] = W[512 + c * 64 + i8 * 8 + j];
  }
  for (int i = tid; i < 3072; i += 32 * WPB) {
    int c = i / 384, rem = i - c * 384;
    int r = rem >> 3, j = rem & 7;
    int t = r >> 3, ii = r & 7;
    Wvf[i] = W[1024 + t * 512 + c * 64 + ii * 8 + j];
  }
  __syncthreads();

  const int m = lane;
  const size_t pair = (size_t)v * 32 + m;
  const int   nbr = nbrs[pair];
  const int   msk = mask[pair];
  const float u0 = rpu[pair * 2 + 0];
  const float u1 = rpu[pair * 2 + 1];

  // ---- gather neighbor features (overlaps with the async stream) -------------
  float xn[64];
  {
    const float4* xp = (const float4*)(x + (size_t)nbr * 64);
#pragma unroll
    for (int q = 0; q < 16; ++q) {
      float4 t4 = xp[q];
      xn[q * 4 + 0] = t4.x; xn[q * 4 + 1] = t4.y;
      xn[q * 4 + 2] = t4.z; xn[q * 4 + 3] = t4.w;
    }
  }

  // ---- Q for this vertex (lanes 0..7 compute one row each) -------------------
  if (lane < 8) {
    const int i = lane;
    float acc = 0.f;
    const float* xv = x + (size_t)v * 64;
#pragma unroll
    for (int c = 0; c < 8; ++c)
#pragma unroll
      for (int j = 0; j < 8; ++j)
        acc += WQf[c * 64 + i * 8 + j] * xv[c * 8 + j];
    Qls[wave][i] = acc;
  }

  // ---- wait for the async P stream, then parallel transport ------------------
  asm volatile("s_wait_asynccnt 0x0" ::: "memory");
  {
    const float* pr = &Scr[wave][m * PSTRIDE];
    float* fw = &Ff[wave][m * 64];
#pragma unroll
    for (int i = 0; i < 8; ++i) {
      float4 p0 = *(const float4*)(pr + i * 8);
      float4 p1 = *(const float4*)(pr + i * 8 + 4);
#pragma unroll
      for (int c = 0; c < 8; ++c) {
        const float* xc = xn + c * 8;
        float fv = p0.x * xc[0] + p0.y * xc[1] + p0.z * xc[2] + p0.w * xc[3] +
                   p1.x * xc[4] + p1.y * xc[5] + p1.z * xc[6] + p1.w * xc[7];
        fw[c * 8 + i] = fv;   // f[c,i] of this pair
      }
    }
  }
  __syncthreads();

  // ---- GEMM-K: K[i,m] = sum_{c,j} WK[c,i,j] f[c,j,m]; K-dim = 64 = 16 chunks -
  {
    v2f af[16];
#pragma unroll
    for (int kc = 0; kc < 16; ++kc) af[kc] = f2_zero();
    if (lo < 8) {  // rows 8..15 of the A tile are zero padding
      const float* base = &WKf[lo * 64 + 2 * hi];
#pragma unroll
      for (int kc = 0; kc < 16; ++kc)
        af[kc] = *(const v2f*)(base + 4 * kc);
    }
#pragma unroll
    for (int nt = 0; nt < 2; ++nt) {
      const float* fb = &Ff[wave][(16 * nt + lo) * 64 + 2 * hi];
      v8f acc0 = f8_zero(), acc1 = f8_zero();
#pragma unroll
      for (int kc = 0; kc < 8; ++kc) {
        v2f b0 = *(const v2f*)(fb + 4 * kc);
        v2f b1 = *(const v2f*)(fb + 4 * (kc + 8));
        acc0 = __builtin_amdgcn_wmma_f32_16x16x4_f32(false, af[kc], false, b0,
                                                     (short)0, acc0, false, false);
        acc1 = __builtin_amdgcn_wmma_f32_16x16x4_f32(false, af[kc + 8], false, b1,
                                                     (short)0, acc1, false, false);
      }
      if (hi == 0) {  // rows 0..7 hold the real K-scores
#pragma unroll
        for (int r = 0; r < 8; ++r)
          Kls[wave][r * 33 + 16 * nt + lo] = acc0[r] + acc1[r];
      }
    }
  }
  __syncthreads();

  // ---- attention -------------------------------------------------------------
  float att;
  {
    float s = 0.f;
#pragma unroll
    for (int i = 0; i < 8; ++i) {
      float t = Qls[wave][i] + Kls[wave][i * 33 + m];
      s += fmaxf(t, 0.f);
    }
    s *= 0.125f;
    s = msk ? s : 0.f;
    float tot = wave_sum32(s);
    att = s / fmaxf(tot, 1e-8f);
  }

  const float q0 = u0 * u0;
  const float q1 = 2.f * u0 * u1;
  const float q2 = u1 * u1;

  // ---- per channel: value GEMM (12 fully dense WMMAs) + combine + reduce -----
  float* Vdw = &Scr[wave][0];   // Vd[48][33] view (P staging is dead by now)
  for (int c = 0; c < 8; ++c) {
    v2f bf[2][2];  // [nt][kc] : B fragments, K = 4*kc + 2*hi + e  (all 8 K used)
#pragma unroll
    for (int nt = 0; nt < 2; ++nt) {
      const float* fb = &Ff[wave][(16 * nt + lo) * 64 + c * 8 + 2 * hi];
      bf[nt][0] = *(const v2f*)(fb);
      bf[nt][1] = *(const v2f*)(fb + 4);
    }
#pragma unroll
    for (int mt = 0; mt < 3; ++mt) {  // 48 rows = 3 full M tiles, no padding
      const float* ab = &Wvf[(c * 48 + 16 * mt + lo) * 8 + 2 * hi];
      v2f a0 = *(const v2f*)(ab);
      v2f a1 = *(const v2f*)(ab + 4);
#pragma unroll
      for (int nt = 0; nt < 2; ++nt) {
        v8f d = f8_zero();
        d = __builtin_amdgcn_wmma_f32_16x16x4_f32(false, a0, false, bf[nt][0],
                                                  (short)0, d, false, false);
        d = __builtin_amdgcn_wmma_f32_16x16x4_f32(false, a1, false, bf[nt][1],
                                                  (short)0, d, false, false);
#pragma unroll
        for (int r8 = 0; r8 < 8; ++r8)
          Vdw[(16 * mt + 8 * hi + r8) * 33 + 16 * nt + lo] = d[r8];
      }
    }
    __syncthreads();

    // combine Taylor orders for this lane's pair, weight by attention, reduce over m
#pragma unroll
    for (int i = 0; i < 8; ++i) {
      float val = Vdw[(i)      * 33 + m]
                + u0 * Vdw[(8  + i) * 33 + m]
                + u1 * Vdw[(16 + i) * 33 + m]
                + q0 * Vdw[(24 + i) * 33 + m]
                + q1 * Vdw[(32 + i) * 33 + m]
                + q2 * Vdw[(40 + i) * 33 + m];
      float r = wave_sum32(att * val);
      if (lane == i) out[(size_t)v * 64 + c * 8 + i] = r;
    }
    __syncthreads();
  }
}

// ---------------------------------------------------------------------------
extern "C" void kernel_launch(void* const* d_in, const int* in_sizes, int n_in,
                              void* d_out, int out_size, void* d_ws, size_t ws_size,
                              hipStream_t stream) {
  (void)n_in; (void)out_size; (void)ws_size;
  const float* x    = (const float*)d_in[0];
  const int*   nbrs = (const int*)d_in[1];
  const int*   mask = (const int*)d_in[2];   // bool in reference; harness passes ints
  const float* ptm  = (const float*)d_in[3];
  const float* rpu  = (const float*)d_in[4];
  const float* rtr  = (const float*)d_in[5];
  const float* qc   = (const float*)d_in[6];
  const float* kc   = (const float*)d_in[7];
  const float* vb0  = (const float*)d_in[8];
  const float* vb1  = (const float*)d_in[9];
  const float* vb2  = (const float*)d_in[10];
  const float* zp   = (const float*)d_in[11];
  const float* fp   = (const float*)d_in[12];
  const float* sp   = (const float*)d_in[13];
  float* out = (float*)d_out;
  float* W   = (float*)d_ws;                 // 4096 f32 = 16 KB scratch

  const int nv = in_sizes[0] / 64;           // [V,C,N] with C=N=8

  ge_prep_weights<<<16, 256, 0, stream>>>(rtr, qc, kc, vb0, vb1, vb2, zp, fp, sp, W);
  const int blocks = (nv + WPB - 1) / WPB;
  ge_attn<<<blocks, 32 * WPB, 0, stream>>>(x, nbrs, mask, ptm, rpu, W, out, nv);
}